// ModalitySpecificLocalSelfAttentionlayer_30657476559024
// MI455X (gfx1250) — compile-verified
//
#include <hip/hip_runtime.h>
#include <hip/hip_bf16.h>

typedef __attribute__((ext_vector_type(2))) float v2f;
typedef __attribute__((ext_vector_type(4))) float v4f;
typedef __attribute__((ext_vector_type(8))) float v8f;

#define CCH 128
#define HDIM 128
#define WDIM 128
#define HW (HDIM * WDIM)
#define KSZ 7
#define KK2 49
#define TPITCH 132          // LDS tile pitch in dwords (bank-conflict-free pad)

// ---------------------------------------------------------------------------
// 1x1-conv (GEMM) + folded BN (+optional ReLU) with fp32 WMMA.
//   Y[o,p] = act( scale[o] * (sum_c W[o,c] * X[c,p]) + bias[o] )
// Block = 256 thr (8 waves) computes all 128 out-channels for 64 pixels.
// B tile (64 px x 128 K) is staged once into LDS (shared by all 8 waves),
// stored transposed tile[p][k] so each WMMA B fragment is one ds_load_b64.
// Optional second source Xb supplies channels [128,256) for the concat conv.
// ---------------------------------------------------------------------------
__global__ __launch_bounds__(256) void gemm_bn_wmma(
    const float* __restrict__ Xa, long sCa, long sPa,
    const float* __restrict__ Xb, long sCb, long sPb,
    const float* __restrict__ Wm, int ldw,
    const float* __restrict__ scale, const float* __restrict__ bias,
    float* __restrict__ Y, long sCo, long sPo, int relu)
{
    __shared__ float tile[64 * TPITCH];    // 33,792 B

    const int tid  = threadIdx.x;
    const int lane = tid & 31;
    const int wv   = tid >> 5;
    const int hl   = lane >> 4;            // half-wave (K pair select)
    const int l16  = lane & 15;
    const int m0   = wv << 4;              // out-channel tile base
    const long p0  = (long)blockIdx.x * 64;

    v8f acc[4] = {};

    const float* srcX[2] = {Xa, Xb};
    const long   sCs[2]  = {sCa, sCb};
    const long   sPs[2]  = {sPa, sPb};
    const int nseg = (Xb != nullptr) ? 2 : 1;

    // Per-wave LDS read bases for the 4 pixel sub-tiles (dword indices).
    int bbase[4];
    #pragma unroll
    for (int s = 0; s < 4; ++s)
        bbase[s] = ((s << 4) + l16) * TPITCH + (hl << 1);

    for (int seg = 0; seg < nseg; ++seg) {
        const float* __restrict__ X = srcX[seg];
        const long sC = sCs[seg], sP = sPs[seg];

        __syncthreads();                   // tile reuse fence (no-op cost seg 0)
        if (sP == 1) {
            // channel-major source: rows (c) contiguous along p
            #pragma unroll
            for (int i = 0; i < 8; ++i) {
                const int e  = tid + (i << 8);       // 0..2047 float4 groups
                const int p4 = (e & 15) << 2;
                const int c  = e >> 4;
                const v4f xv = *(const v4f*)(X + (long)c * sC + p0 + p4);
                tile[(p4 + 0) * TPITCH + c] = xv.x;
                tile[(p4 + 1) * TPITCH + c] = xv.y;
                tile[(p4 + 2) * TPITCH + c] = xv.z;
                tile[(p4 + 3) * TPITCH + c] = xv.w;
            }
        } else {
            // pixel-major source: channels contiguous per pixel
            #pragma unroll
            for (int i = 0; i < 8; ++i) {
                const int e  = tid + (i << 8);
                const int c4 = (e & 31) << 2;
                const int p  = e >> 5;
                const v4f xv = *(const v4f*)(X + (p0 + p) * sP + c4);
                *(v4f*)(&tile[p * TPITCH + c4]) = xv;
            }
        }
        __syncthreads();

        // Per-lane A pointer: row (m0+l16) of this 128-wide K segment.
        const float* __restrict__ Ap =
            Wm + (long)(m0 + l16) * ldw + (seg << 7) + (hl << 1);

        #pragma unroll
        for (int k0 = 0; k0 < CCH; k0 += 4) {
            const v2f a = *(const v2f*)(Ap + k0);          // global b64
            #pragma unroll
            for (int s = 0; s < 4; ++s) {
                const v2f b = *(const v2f*)(&tile[bbase[s] + k0]); // ds b64
                acc[s] = __builtin_amdgcn_wmma_f32_16x16x4_f32(
                    false, a, false, b, (short)0, acc[s], false, false);
            }
        }
    }

    // Epilogue: BN scale/bias (+ReLU). D layout: VGPR r -> M = r + 8*half.
    #pragma unroll
    for (int s = 0; s < 4; ++s) {
        const long p = p0 + (s << 4) + l16;
        #pragma unroll
        for (int r = 0; r < 8; ++r) {
            const int o = m0 + r + (hl << 3);
            float f = acc[s][r] * scale[o] + bias[o];
            if (relu) f = fmaxf(f, 0.0f);
            Y[(long)o * sCo + p * sPo] = f;
        }
    }
}

// ---------------------------------------------------------------------------
// 7x7 windowed self-attention. q,k,v,nf are pixel-major: t[p*128 + c].
// One wave per pixel; lane owns channels 4*lane..4*lane+3 (contiguous b128).
// Zero-padding semantics preserved: OOB neighbor => sim = 0 logit (still in
// softmax), zero v contribution.
// ---------------------------------------------------------------------------
__global__ __launch_bounds__(256) void local_attn_kernel(
    const float* __restrict__ q, const float* __restrict__ k,
    const float* __restrict__ v, float* __restrict__ nf)
{
    const int lane = threadIdx.x & 31;
    const int wv   = threadIdx.x >> 5;
    const int p    = blockIdx.x * 8 + wv;
    const int h    = p >> 7;               // H = 128
    const int w    = p & (WDIM - 1);

    const v4f qv = *(const v4f*)(q + (long)p * CCH + lane * 4);

    float sim[KK2];
    #pragma unroll
    for (int t = 0; t < KK2; ++t) {
        const int hn = h + t / KSZ - (KSZ / 2);
        const int wn = w + t % KSZ - (KSZ / 2);
        float d = 0.0f;
        if ((unsigned)hn < (unsigned)HDIM && (unsigned)wn < (unsigned)WDIM) {
            const long pn = (long)hn * WDIM + wn;
            const v4f kv = *(const v4f*)(k + pn * CCH + lane * 4);
            d = qv.x * kv.x + qv.y * kv.y + qv.z * kv.z + qv.w * kv.w;
        }
        #pragma unroll
        for (int off = 16; off > 0; off >>= 1)
            d += __shfl_xor(d, off, 32);
        sim[t] = d;
    }

    float m = sim[0];
    #pragma unroll
    for (int t = 1; t < KK2; ++t) m = fmaxf(m, sim[t]);
    float z = 0.0f;
    #pragma unroll
    for (int t = 0; t < KK2; ++t) { sim[t] = __expf(sim[t] - m); z += sim[t]; }
    const float inv = 1.0f / z;

    v4f acc = {0.0f, 0.0f, 0.0f, 0.0f};
    #pragma unroll
    for (int t = 0; t < KK2; ++t) {
        const int hn = h + t / KSZ - (KSZ / 2);
        const int wn = w + t % KSZ - (KSZ / 2);
        if ((unsigned)hn < (unsigned)HDIM && (unsigned)wn < (unsigned)WDIM) {
            const long pn = (long)hn * WDIM + wn;
            const v4f vv = *(const v4f*)(v + pn * CCH + lane * 4);
            const float wt = sim[t] * inv;
            acc.x += wt * vv.x; acc.y += wt * vv.y;
            acc.z += wt * vv.z; acc.w += wt * vv.w;
        }
    }
    *(v4f*)(nf + (long)p * CCH + lane * 4) = acc;
}

extern "C" void kernel_launch(void* const* d_in, const int* in_sizes, int n_in,
                              void* d_out, int out_size, void* d_ws, size_t ws_size,
                              hipStream_t stream) {
    const float* x    = (const float*)d_in[0];
    const float* w_q1 = (const float*)d_in[1];
    const float* s_q1 = (const float*)d_in[2];
    const float* b_q1 = (const float*)d_in[3];
    const float* w_q2 = (const float*)d_in[4];
    const float* s_q2 = (const float*)d_in[5];
    const float* b_q2 = (const float*)d_in[6];
    const float* w_k1 = (const float*)d_in[7];
    const float* s_k1 = (const float*)d_in[8];
    const float* b_k1 = (const float*)d_in[9];
    const float* w_k2 = (const float*)d_in[10];
    const float* s_k2 = (const float*)d_in[11];
    const float* b_k2 = (const float*)d_in[12];
    const float* w_v  = (const float*)d_in[13];
    const float* s_v  = (const float*)d_in[14];
    const float* b_v  = (const float*)d_in[15];
    const float* w_o  = (const float*)d_in[16];
    const float* s_o  = (const float*)d_in[17];
    const float* b_o  = (const float*)d_in[18];
    float* out = (float*)d_out;

    // Pixel-major scratch buffers (HW x C each) in workspace.
    const size_t NB = (size_t)HW * CCH;
    float* q1 = (float*)d_ws;
    float* k1 = q1 + NB;
    float* vb = k1 + NB;
    float* qb = vb + NB;
    float* kb = qb + NB;
    float* nf = kb + NB;

    const dim3 blk(256);
    const dim3 grd(HW / 64);

    // Stage 1: three independent C x C GEMMs from channel-major x.
    gemm_bn_wmma<<<grd, blk, 0, stream>>>(x, HW, 1, nullptr, 0, 0,
                                          w_q1, CCH, s_q1, b_q1,
                                          q1, 1, CCH, 1);
    gemm_bn_wmma<<<grd, blk, 0, stream>>>(x, HW, 1, nullptr, 0, 0,
                                          w_k1, CCH, s_k1, b_k1,
                                          k1, 1, CCH, 1);
    gemm_bn_wmma<<<grd, blk, 0, stream>>>(x, HW, 1, nullptr, 0, 0,
                                          w_v, CCH, s_v, b_v,
                                          vb, 1, CCH, 1);
    // Stage 2: second projections from pixel-major intermediates.
    gemm_bn_wmma<<<grd, blk, 0, stream>>>(q1, 1, CCH, nullptr, 0, 0,
                                          w_q2, CCH, s_q2, b_q2,
                                          qb, 1, CCH, 1);
    gemm_bn_wmma<<<grd, blk, 0, stream>>>(k1, 1, CCH, nullptr, 0, 0,
                                          w_k2, CCH, s_k2, b_k2,
                                          kb, 1, CCH, 1);
    // Stage 3: 7x7 local attention (one wave per pixel).
    local_attn_kernel<<<dim3(HW / 8), blk, 0, stream>>>(qb, kb, vb, nf);
    // Stage 4: final conv over concat([nf, x]) -> channel-major output, no ReLU.
    gemm_bn_wmma<<<grd, blk, 0, stream>>>(nf, 1, CCH, x, HW, 1,
                                          w_o, 2 * CCH, s_o, b_o,
                                          out, HW, 1, 0);
}